// CIM_16011638079733
// MI455X (gfx1250) — compile-verified
//
#include <hip/hip_runtime.h>
#include <math.h>

#define B_SZ   2
#define DIMC   128
#define D_INNER 256
#define D_STATE 16
#define HW     4096      // 64*64
#define NT     8192      // B_SZ*HW tokens
#define CMID   42        // DIM//3

typedef float v2f __attribute__((ext_vector_type(2)));
typedef float v8f __attribute__((ext_vector_type(8)));

__device__ __forceinline__ float dev_silu(float x)    { return x / (1.f + __expf(-x)); }
__device__ __forceinline__ float dev_sigmoid(float x) { return 1.f / (1.f + __expf(-x)); }

// ---------------------------------------------------------------------------
// fp32 WMMA GEMM, 32x32 macro-tile per wave (2x2 sub-tiles of 16x16):
//   C[M,N] = epi( A[M,K] * B + bias[N] ) + resid
// BL==0: weights stored [N,K] row-major (einsum '...k,nk->...n'), elem n*ldb+k
// BL==1: weights stored [K,N] row-major (plain X@W),              elem k*ldb+n
// EPI: 0=none 1=relu 2=sigmoid 3=softplus
// Per K-step: 2 A-frags + 2 B-frags -> 4 v_wmma_f32_16x16x4_f32 (2x reuse each).
// ---------------------------------------------------------------------------
template<int EPI, int BL>
__global__ __launch_bounds__(128)
void gemm32_wmma(const float* __restrict__ A, int lda,
                 const float* __restrict__ Bw, int ldb,
                 const float* __restrict__ bias,
                 const float* __restrict__ resid, int ldr,
                 float* __restrict__ C, int ldc,
                 int M, int N, int K)
{
    const int lane = threadIdx.x;          // 0..31
    const int half = lane >> 4;            // 0/1
    const int lm   = lane & 15;
    const int mt   = blockIdx.y * blockDim.y + threadIdx.y;
    const int m0   = mt * 32;
    const int n0   = blockIdx.x * 32;
    if (m0 >= M) return;                   // uniform across wave

    const int  na0  = n0 + lm;
    const int  na1  = n0 + 16 + lm;
    const bool nok0 = (na0 < N);
    const bool nok1 = (na1 < N);
    const int  ns0  = nok0 ? na0 : (N - 1);   // clamp addr, zero via select
    const int  ns1  = nok1 ? na1 : (N - 1);

    const float* arow0 = A + (size_t)(m0 + lm) * lda + 2 * half;
    const float* arow1 = arow0 + (size_t)16 * lda;
    const float* brow0 = nullptr;
    const float* brow1 = nullptr;
    if (BL == 0) {
        brow0 = Bw + (size_t)ns0 * ldb + 2 * half;
        brow1 = Bw + (size_t)ns1 * ldb + 2 * half;
    }

    v8f acc00 = {0.f,0.f,0.f,0.f,0.f,0.f,0.f,0.f};
    v8f acc01 = acc00, acc10 = acc00, acc11 = acc00;

    for (int k0 = 0; k0 < K; k0 += 4) {
        v2f a0, a1, b0, b1;
        a0[0] = arow0[k0];  a0[1] = arow0[k0 + 1];
        a1[0] = arow1[k0];  a1[1] = arow1[k0 + 1];
        if (BL == 0) {
            float p0 = brow0[k0], p1 = brow0[k0 + 1];
            float q0 = brow1[k0], q1 = brow1[k0 + 1];
            b0[0] = nok0 ? p0 : 0.f;  b0[1] = nok0 ? p1 : 0.f;
            b1[0] = nok1 ? q0 : 0.f;  b1[1] = nok1 ? q1 : 0.f;
        } else {
            int k = k0 + 2 * half;
            float p0 = Bw[(size_t)k * ldb + ns0];
            float p1 = Bw[(size_t)(k + 1) * ldb + ns0];
            float q0 = Bw[(size_t)k * ldb + ns1];
            float q1 = Bw[(size_t)(k + 1) * ldb + ns1];
            b0[0] = nok0 ? p0 : 0.f;  b0[1] = nok0 ? p1 : 0.f;
            b1[0] = nok1 ? q0 : 0.f;  b1[1] = nok1 ? q1 : 0.f;
        }
        acc00 = __builtin_amdgcn_wmma_f32_16x16x4_f32(false, a0, false, b0, (short)0, acc00, false, false);
        acc01 = __builtin_amdgcn_wmma_f32_16x16x4_f32(false, a0, false, b1, (short)0, acc01, false, false);
        acc10 = __builtin_amdgcn_wmma_f32_16x16x4_f32(false, a1, false, b0, (short)0, acc10, false, false);
        acc11 = __builtin_amdgcn_wmma_f32_16x16x4_f32(false, a1, false, b1, (short)0, acc11, false, false);
    }

    const float bv0 = (bias != nullptr && nok0) ? bias[na0] : 0.f;
    const float bv1 = (bias != nullptr && nok1) ? bias[na1] : 0.f;

#pragma unroll
    for (int mi = 0; mi < 2; ++mi) {
        const v8f& cA = mi ? acc10 : acc00;   // n-subtile 0
        const v8f& cB = mi ? acc11 : acc01;   // n-subtile 1
#pragma unroll
        for (int v = 0; v < 8; ++v) {
            int m = m0 + mi * 16 + v + 8 * half;
            if (nok0) {
                float x = cA[v] + bv0;
                if      (EPI == 1) x = x > 0.f ? x : 0.f;
                else if (EPI == 2) x = dev_sigmoid(x);
                else if (EPI == 3) x = (x > 20.f) ? x : log1pf(__expf(x));
                if (resid) x += resid[(size_t)m * ldr + na0];
                C[(size_t)m * ldc + na0] = x;
            }
            if (nok1) {
                float x = cB[v] + bv1;
                if      (EPI == 1) x = x > 0.f ? x : 0.f;
                else if (EPI == 2) x = dev_sigmoid(x);
                else if (EPI == 3) x = (x > 20.f) ? x : log1pf(__expf(x));
                if (resid) x += resid[(size_t)m * ldr + na1];
                C[(size_t)m * ldc + na1] = x;
            }
        }
    }
}

// NCHW [B,128,4096] -> token-major [B*4096,128]
__global__ void k_nchw_to_tok(const float* __restrict__ x, float* __restrict__ f)
{
    int idx = blockIdx.x * blockDim.x + threadIdx.x;   // b*128*4096 flat
    if (idx >= B_SZ * DIMC * HW) return;
    int hw = idx & (HW - 1);
    int c  = (idx >> 12) & (DIMC - 1);
    int b  = idx >> 19;
    f[((size_t)(b * HW + hw)) * DIMC + c] = x[idx];
}

// LayerNorm of both streams + token-exchange (x2n == ln2(f2); x1n mixes)
__global__ __launch_bounds__(256)
void k_ln_swap(const float* __restrict__ f1, const float* __restrict__ f2,
               const float* __restrict__ w1, const float* __restrict__ b1,
               const float* __restrict__ w2, const float* __restrict__ b2,
               float* __restrict__ x1n, float* __restrict__ x2n)
{
    int t    = blockIdx.x * blockDim.y + threadIdx.y;   // token
    int lane = threadIdx.x;
    const size_t base = (size_t)t * DIMC;
    float a[4], c2[4];
    float s1 = 0.f, q1 = 0.f, s2 = 0.f, q2 = 0.f;
#pragma unroll
    for (int i = 0; i < 4; ++i) {
        int c = lane + 32 * i;
        a[i]  = f1[base + c];  c2[i] = f2[base + c];
        s1 += a[i];  q1 += a[i]*a[i];
        s2 += c2[i]; q2 += c2[i]*c2[i];
    }
#pragma unroll
    for (int off = 16; off > 0; off >>= 1) {
        s1 += __shfl_xor(s1, off, 32); q1 += __shfl_xor(q1, off, 32);
        s2 += __shfl_xor(s2, off, 32); q2 += __shfl_xor(q2, off, 32);
    }
    float m1 = s1 * (1.f/DIMC), v1 = q1 * (1.f/DIMC) - m1*m1;
    float m2 = s2 * (1.f/DIMC), v2 = q2 * (1.f/DIMC) - m2*m2;
    float r1 = rsqrtf(v1 + 1e-5f), r2 = rsqrtf(v2 + 1e-5f);
#pragma unroll
    for (int i = 0; i < 4; ++i) {
        int c = lane + 32 * i;
        float l1 = (a[i]  - m1) * r1 * w1[c] + b1[c];
        float l2 = (c2[i] - m2) * r2 * w2[c] + b2[c];
        x1n[base + c] = (c < 79) ? l2 : l1;
        x2n[base + c] = l2;
    }
}

// causal depthwise conv width-4 along L (+bias, silu). xz:[NT,512] -> xc:[NT,256]
__global__ void k_mamba_conv(const float* __restrict__ xz,
                             const float* __restrict__ cw,
                             const float* __restrict__ cb,
                             float* __restrict__ xc)
{
    int idx = blockIdx.x * blockDim.x + threadIdx.x;   // NT*256
    if (idx >= NT * D_INNER) return;
    int d = idx & (D_INNER - 1);
    int t = idx >> 8;
    int l = t & (HW - 1);
    float acc = cb[d];
#pragma unroll
    for (int k = 0; k < 4; ++k) {
        int ll = l - 3 + k;
        if (ll >= 0) acc += cw[d * 4 + k] * xz[((size_t)(t - 3 + k)) * 512 + d];
    }
    xc[(size_t)t * D_INNER + d] = dev_silu(acc);
}

// selective scan: one thread per (b,d); B/C staged through LDS in 64-step chunks.
// Fuses +u*D and *silu(z) epilogue.
__global__ __launch_bounds__(256)
void k_mamba_scan(const float* __restrict__ xz,  const float* __restrict__ xc,
                  const float* __restrict__ dt,  const float* __restrict__ dbl,
                  const float* __restrict__ A_log, const float* __restrict__ Dp,
                  float* __restrict__ y)
{
    const int b = blockIdx.x;
    const int d = threadIdx.x;
    __shared__ float Bs[64][D_STATE];
    __shared__ float Cs[64][D_STATE];
    float An[D_STATE], h[D_STATE];
#pragma unroll
    for (int n = 0; n < D_STATE; ++n) { An[n] = -__expf(A_log[d * D_STATE + n]); h[n] = 0.f; }
    const float Dd = Dp[d];

    for (int l0 = 0; l0 < HW; l0 += 64) {
        __syncthreads();
        for (int i = threadIdx.x; i < 64 * 32; i += 256) {
            int r = i >> 5, col = i & 31;
            float v = dbl[((size_t)(b * HW + l0 + r)) * 40 + 8 + col];
            if (col < 16) Bs[r][col] = v; else Cs[r][col - 16] = v;
        }
        __syncthreads();
        for (int r = 0; r < 64; ++r) {
            size_t t  = (size_t)(b * HW + l0 + r);
            float dtv = dt[t * D_INNER + d];
            float u   = xc[t * D_INNER + d];
            float du  = dtv * u;
            float yv  = 0.f;
#pragma unroll
            for (int n = 0; n < D_STATE; ++n) {
                h[n] = __expf(dtv * An[n]) * h[n] + du * Bs[r][n];
                yv  += h[n] * Cs[r][n];
            }
            yv += u * Dd;
            float z = xz[t * 512 + D_INNER + d];
            y[t * D_INNER + d] = yv * dev_silu(z);
        }
    }
}

__global__ void k_concat(const float* __restrict__ f1, const float* __restrict__ f2,
                         float* __restrict__ g)
{
    int idx = blockIdx.x * blockDim.x + threadIdx.x;   // NT*256
    if (idx >= NT * 256) return;
    int c = idx & 255, t = idx >> 8;
    g[idx] = (c < DIMC) ? f1[(size_t)t * DIMC + c] : f2[(size_t)t * DIMC + c - DIMC];
}

// direct 3x3 conv, pad 1. act: 0 = none, 1 = exact gelu
__global__ void k_conv3x3(const float* __restrict__ in, const float* __restrict__ w,
                          const float* __restrict__ bias, float* __restrict__ out,
                          int Cin, int Cout, int act, int total)
{
    int idx = blockIdx.x * blockDim.x + threadIdx.x;
    if (idx >= total) return;
    int hw = idx & (HW - 1);
    int t  = idx >> 12;
    int co = t % Cout;
    int b  = t / Cout;
    int yy = hw >> 6, xx = hw & 63;
    float acc = bias ? bias[co] : 0.f;
    for (int ci = 0; ci < Cin; ++ci) {
        const float* ip = in + ((size_t)(b * Cin + ci)) * HW;
        const float* wp = w + ((size_t)(co * Cin + ci)) * 9;
#pragma unroll
        for (int ky = 0; ky < 3; ++ky) {
            int iy = yy + ky - 1;
            if (iy < 0 || iy > 63) continue;
#pragma unroll
            for (int kx = 0; kx < 3; ++kx) {
                int ix = xx + kx - 1;
                if (ix < 0 || ix > 63) continue;
                acc += wp[ky * 3 + kx] * ip[iy * 64 + ix];
            }
        }
    }
    if (act == 1) acc = 0.5f * acc * (1.f + erff(acc * 0.70710678f));
    out[idx] = acc;
}

__global__ __launch_bounds__(256)
void k_pool_avgmax(const float* __restrict__ y, float* __restrict__ avg, float* __restrict__ mx)
{
    int bc = blockIdx.x;                 // b*128 + ch
    const float* p = y + (size_t)bc * HW;
    float s = 0.f, m = -3.4e38f;
    for (int i = threadIdx.x; i < HW; i += 256) { float v = p[i]; s += v; m = fmaxf(m, v); }
    __shared__ float ss[256], sm[256];
    ss[threadIdx.x] = s; sm[threadIdx.x] = m;
    __syncthreads();
    for (int o = 128; o > 0; o >>= 1) {
        if (threadIdx.x < o) {
            ss[threadIdx.x] += ss[threadIdx.x + o];
            sm[threadIdx.x]  = fmaxf(sm[threadIdx.x], sm[threadIdx.x + o]);
        }
        __syncthreads();
    }
    if (threadIdx.x == 0) { avg[bc] = ss[0] * (1.f / HW); mx[bc] = sm[0]; }
}

__global__ __launch_bounds__(128)
void k_cab_attn(const float* __restrict__ avg, const float* __restrict__ mx,
                const float* __restrict__ fc1, const float* __restrict__ fc2,
                float* __restrict__ attn)
{
    int b = blockIdx.x, tid = threadIdx.x;
    __shared__ float hA[8], hM[8];
    if (tid < 8) {
        float s = 0.f;
        for (int c = 0; c < DIMC; ++c) s += fc1[tid * DIMC + c] * avg[b * DIMC + c];
        hA[tid] = dev_silu(s);
    } else if (tid < 16) {
        int j = tid - 8;
        float s = 0.f;
        for (int c = 0; c < DIMC; ++c) s += fc1[j * DIMC + c] * mx[b * DIMC + c];
        hM[j] = dev_silu(s);
    }
    __syncthreads();
    float s = 0.f;
#pragma unroll
    for (int j = 0; j < 8; ++j) s += fc2[tid * 8 + j] * (hA[j] + hM[j]);
    attn[b * DIMC + tid] = dev_sigmoid(s);
}

__global__ void k_sam(const float* __restrict__ x1, const float* __restrict__ x2,
                      const float* __restrict__ w1, const float* __restrict__ b1,
                      const float* __restrict__ w2, const float* __restrict__ b2,
                      float* __restrict__ wout)
{
    int idx = blockIdx.x * blockDim.x + threadIdx.x;   // B*HW
    if (idx >= B_SZ * HW) return;
    int hw = idx & (HW - 1);
    int b  = idx >> 12;
    float s1 = 0.f, m1 = -3.4e38f, s2 = 0.f, m2 = -3.4e38f;
    for (int c = 0; c < DIMC; ++c) {
        float v1 = x1[((size_t)(b * DIMC + c)) * HW + hw];
        float v2 = x2[((size_t)(b * DIMC + c)) * HW + hw];
        s1 += v1; m1 = fmaxf(m1, v1);
        s2 += v2; m2 = fmaxf(m2, v2);
    }
    float f[4] = { s1 * (1.f/DIMC), m1, s2 * (1.f/DIMC), m2 };
    float o0 = b2[0], o1 = b2[1];
#pragma unroll
    for (int j = 0; j < 16; ++j) {
        float h = b1[j];
#pragma unroll
        for (int i = 0; i < 4; ++i) h += w1[j * 4 + i] * f[i];
        h = h > 0.f ? h : 0.f;
        o0 += w2[j]      * h;
        o1 += w2[16 + j] * h;
    }
    wout[b * 2 * HW + hw]      = dev_sigmoid(o0);
    wout[b * 2 * HW + HW + hw] = dev_sigmoid(o1);
}

__global__ void k_final(const float* __restrict__ x1, const float* __restrict__ x2,
                        const float* __restrict__ g,
                        const float* __restrict__ cy1, const float* __restrict__ cy2,
                        const float* __restrict__ attn1, const float* __restrict__ attn2,
                        const float* __restrict__ wsam, float* __restrict__ out)
{
    int idx = blockIdx.x * blockDim.x + threadIdx.x;   // B*128*HW
    if (idx >= B_SZ * DIMC * HW) return;
    int hw = idx & (HW - 1);
    int c  = (idx >> 12) & (DIMC - 1);
    int b  = idx >> 19;
    float gv  = g[((size_t)(b * HW + hw)) * DIMC + c];
    float xv1 = x1[idx], xv2 = x2[idx];
    float attv1 = cy1[idx] * attn1[b * DIMC + c] + wsam[b * 2 * HW + hw]      * xv1;
    float attv2 = cy2[idx] * attn2[b * DIMC + c] + wsam[b * 2 * HW + HW + hw] * xv2;
    out[idx]                      = xv1 + (1.f - gv) * attv2;
    out[B_SZ * DIMC * HW + idx]   = xv2 + gv * attv1;
}

// ---------------------------------------------------------------------------
struct MambaP { const float *inw,*cw,*cb,*xpw,*dtw,*dtb,*alog,*Dp,*outw; };
struct TsmP   { const float *l1w,*l1b,*l2w,*l2b; MambaP m1, m2; };

extern "C" void kernel_launch(void* const* d_in, const int* in_sizes, int n_in,
                              void* d_out, int out_size, void* d_ws, size_t ws_size,
                              hipStream_t stream)
{
    (void)in_sizes; (void)n_in; (void)out_size; (void)ws_size;
    int ii = 0;
    auto F = [&]() { return (const float*)d_in[ii++]; };

    const float* x1 = F();
    const float* x2 = F();
    TsmP ts[2];
    for (int t = 0; t < 2; ++t) {
        ts[t].l1w = F(); ts[t].l1b = F(); ts[t].l2w = F(); ts[t].l2b = F();
        MambaP* ms[2] = { &ts[t].m1, &ts[t].m2 };
        for (int m = 0; m < 2; ++m) {
            MambaP* p = ms[m];
            p->inw = F(); p->cw = F(); p->cb = F(); p->xpw = F(); p->dtw = F();
            p->dtb = F(); p->alog = F(); p->Dp = F(); p->outw = F();
        }
    }
    struct CabP { const float *w1,*b1,*w2,*b2,*fc1,*fc2; } cab[2];
    for (int i = 0; i < 2; ++i) {
        cab[i].w1 = F(); cab[i].b1 = F(); cab[i].w2 = F();
        cab[i].b2 = F(); cab[i].fc1 = F(); cab[i].fc2 = F();
    }
    const float *sw1 = F(), *sb1 = F(), *sw2 = F(), *sb2 = F();
    const float *gw1 = F(), *gb1 = F(), *gw2 = F(), *gb2 = F();

    // workspace layout (floats)
    float* ws = (float*)d_ws;
    float* f1    = ws;              ws += (size_t)NT * DIMC;
    float* f2    = ws;              ws += (size_t)NT * DIMC;
    float* x1n   = ws;              ws += (size_t)NT * DIMC;
    float* x2n   = ws;              ws += (size_t)NT * DIMC;
    float* xz    = ws;              ws += (size_t)NT * 512;
    float* xc    = ws;              ws += (size_t)NT * D_INNER;
    float* dbl   = ws;              ws += (size_t)NT * 40;
    float* dtb   = ws;              ws += (size_t)NT * D_INNER;
    float* ybuf  = ws;              ws += (size_t)NT * D_INNER;
    float* gcat  = ws;              ws += (size_t)NT * 256;
    float* gbuf1 = ws;              ws += (size_t)NT * 64;
    float* gbuf  = ws;              ws += (size_t)NT * DIMC;
    float* cbuf  = ws;              ws += (size_t)B_SZ * CMID * HW;
    float* cy1   = ws;              ws += (size_t)NT * DIMC;
    float* cy2   = ws;              ws += (size_t)NT * DIMC;
    float* pavg  = ws;              ws += 256;
    float* pmax  = ws;              ws += 256;
    float* attn1 = ws;              ws += 256;
    float* attn2 = ws;              ws += 256;
    float* wsam  = ws;              ws += (size_t)B_SZ * 2 * HW;

    const dim3 gblk(32, 4);
#define GEMM(EPI, BL, A, lda, Bw, ldb, bias, res, ldr, C, ldc, M, N, K)                   \
    gemm32_wmma<EPI, BL><<<dim3(((N) + 31) / 32, ((M) + 127) / 128), gblk, 0, stream>>>(  \
        A, lda, Bw, ldb, bias, res, ldr, C, ldc, M, N, K)

    // NCHW -> token-major
    k_nchw_to_tok<<<(B_SZ*DIMC*HW + 255)/256, 256, 0, stream>>>(x1, f1);
    k_nchw_to_tok<<<(B_SZ*DIMC*HW + 255)/256, 256, 0, stream>>>(x2, f2);

    auto run_mamba = [&](const float* xin, const MambaP& mp, float* fres) {
        // in_proj: [NT,512] = xin[NT,128] * W[512,128]^T
        GEMM(0, 0, xin, DIMC, mp.inw, DIMC, nullptr, nullptr, 0, xz, 512, NT, 512, DIMC);
        // causal conv + silu
        k_mamba_conv<<<(NT*D_INNER + 255)/256, 256, 0, stream>>>(xz, mp.cw, mp.cb, xc);
        // x_proj: [NT,40] = xc[NT,256] * W[40,256]^T
        GEMM(0, 0, xc, D_INNER, mp.xpw, D_INNER, nullptr, nullptr, 0, dbl, 40, NT, 40, D_INNER);
        // dt_proj + softplus: [NT,256] = dbl[:, :8] * W[256,8]^T + b
        GEMM(3, 0, dbl, 40, mp.dtw, 8, mp.dtb, nullptr, 0, dtb, D_INNER, NT, D_INNER, 8);
        // selective scan (fused +u*D and *silu(z))
        k_mamba_scan<<<B_SZ, 256, 0, stream>>>(xz, xc, dtb, dbl, mp.alog, mp.Dp, ybuf);
        // out_proj + residual into fres
        GEMM(0, 0, ybuf, D_INNER, mp.outw, D_INNER, nullptr, fres, DIMC, fres, DIMC,
             NT, DIMC, D_INNER);
    };

    for (int t = 0; t < 2; ++t) {
        k_ln_swap<<<NT/8, dim3(32, 8), 0, stream>>>(f1, f2,
            ts[t].l1w, ts[t].l1b, ts[t].l2w, ts[t].l2b, x1n, x2n);
        run_mamba(x1n, ts[t].m1, f1);
        run_mamba(x2n, ts[t].m2, f2);
    }

    // gate MLP: sigmoid(relu([f1|f2] @ w1 + b1) @ w2 + b2)
    k_concat<<<(NT*256 + 255)/256, 256, 0, stream>>>(f1, f2, gcat);
    GEMM(1, 1, gcat, 256, gw1, 64, gb1, nullptr, 0, gbuf1, 64, NT, 64, 256);
    GEMM(2, 1, gbuf1, 64, gw2, DIMC, gb2, nullptr, 0, gbuf, DIMC, NT, DIMC, 64);

    // CAB branches on the original NCHW inputs
    const float* xs[2] = { x1, x2 };
    float* cys[2]      = { cy1, cy2 };
    float* attns[2]    = { attn1, attn2 };
    for (int i = 0; i < 2; ++i) {
        int tot1 = B_SZ * CMID * HW, tot2 = B_SZ * DIMC * HW;
        k_conv3x3<<<(tot1 + 255)/256, 256, 0, stream>>>(xs[i], cab[i].w1, cab[i].b1,
                                                        cbuf, DIMC, CMID, 1, tot1);
        k_conv3x3<<<(tot2 + 255)/256, 256, 0, stream>>>(cbuf, cab[i].w2, cab[i].b2,
                                                        cys[i], CMID, DIMC, 0, tot2);
        k_pool_avgmax<<<B_SZ * DIMC, 256, 0, stream>>>(cys[i], pavg, pmax);
        k_cab_attn<<<B_SZ, DIMC, 0, stream>>>(pavg, pmax, cab[i].fc1, cab[i].fc2, attns[i]);
    }

    // SAM spatial attention
    k_sam<<<(B_SZ*HW + 255)/256, 256, 0, stream>>>(x1, x2, sw1, sb1, sw2, sb2, wsam);

    // final blend -> d_out (out1 then out2)
    k_final<<<(B_SZ*DIMC*HW + 255)/256, 256, 0, stream>>>(
        x1, x2, gbuf, cy1, cy2, attn1, attn2, wsam, (float*)d_out);
#undef GEMM
}